// TplinkerHandshakingKernel_3118146257538
// MI455X (gfx1250) — compile-verified
//
#include <hip/hip_runtime.h>
#include <hip/hip_bf16.h>

typedef __attribute__((ext_vector_type(2))) float v2f;
typedef __attribute__((ext_vector_type(4))) float v4f;
typedef __attribute__((ext_vector_type(8))) float v8f;

namespace {
constexpr int Bn    = 8;
constexpr int Lseq  = 192;
constexpr int Hdim  = 768;
constexpr int Mrows = Bn * Lseq;            // 1536
constexpr int Ncols = 2 * Hdim;             // 1536 (A1 | A2)
constexpr int Kdim  = Hdim;                 // 768
constexpr int Pn    = Lseq * (Lseq + 1) / 2; // 18528
}

// ---------------------------------------------------------------------------
// Kernel 1: C[m][n] = X[m][k] * Wt[k][n], f32 WMMA (exact precision).
//   X  = seq_hiddens flattened (1536 x 768, row-major)
//   Wt[k][n] = W[h][half*H + k]  where n = half*H + h, W row-major (768 x 1536)
// Each wave: 16x64 output strip = 4 accumulators sharing one A fragment.
// Block = 256 threads = 8 waves stacked in M (128 rows). Grid = (24, 12).
// Fragment layouts per CDNA5 ISA:
//   A 16x4:  lanes 0-15 hold M=lane,K={k,k+1}; lanes 16-31 hold K={k+2,k+3}
//   B 4x16:  mirrored (lanes hold N, VGPRs hold K)
//   C 16x16: VGPR v -> M = v + (lane>=16)*8, N = lane&15
// ---------------------------------------------------------------------------
__global__ __launch_bounds__(256) void gemm_proj(const float* __restrict__ X,
                                                 const float* __restrict__ W,
                                                 float* __restrict__ C) {
  const int lane = threadIdx.x & 31;
  const int wave = threadIdx.x >> 5;
  const int m0   = (blockIdx.y * 8 + wave) * 16;
  const int n0   = blockIdx.x * 64;           // 64-wide strip, within one half
  const int half = (n0 >= Hdim) ? 1 : 0;
  const int h0   = n0 - half * Hdim;
  const int l15  = lane & 15;
  const int koff = (lane >> 4) * 2;

  const float* xp  = X + (m0 + l15) * Kdim + koff;
  const float* wp0 = W + (h0 + l15 +  0) * Ncols + half * Hdim + koff;
  const float* wp1 = W + (h0 + l15 + 16) * Ncols + half * Hdim + koff;
  const float* wp2 = W + (h0 + l15 + 32) * Ncols + half * Hdim + koff;
  const float* wp3 = W + (h0 + l15 + 48) * Ncols + half * Hdim + koff;

  v8f c0 = {}, c1 = {}, c2 = {}, c3 = {};

#pragma unroll 4
  for (int k = 0; k < Kdim; k += 4) {
    v2f a  = *(const v2f*)(xp  + k);
    v2f b0 = *(const v2f*)(wp0 + k);
    v2f b1 = *(const v2f*)(wp1 + k);
    v2f b2 = *(const v2f*)(wp2 + k);
    v2f b3 = *(const v2f*)(wp3 + k);
    c0 = __builtin_amdgcn_wmma_f32_16x16x4_f32(false, a, false, b0, (short)0, c0, false, false);
    c1 = __builtin_amdgcn_wmma_f32_16x16x4_f32(false, a, false, b1, (short)0, c1, false, false);
    c2 = __builtin_amdgcn_wmma_f32_16x16x4_f32(false, a, false, b2, (short)0, c2, false, false);
    c3 = __builtin_amdgcn_wmma_f32_16x16x4_f32(false, a, false, b3, (short)0, c3, false, false);
  }

  const int srow = m0 + (lane >> 4) * 8;
  const int scol = n0 + l15;
#pragma unroll
  for (int v = 0; v < 8; ++v) {
    float* crow = C + (srow + v) * Ncols + scol;
    crow[0]  = c0[v];
    crow[16] = c1[v];
    crow[32] = c2[v];
    crow[48] = c3[v];
  }
}

// ---------------------------------------------------------------------------
// Kernel 2: out[b, p(i,j), h] = tanh(A1[b,i,h] + A2[b,j,h] + bias[h]).
// Grid = (i=192, b=8); block = 192 threads; thread t owns h in [4t, 4t+4).
// base = A1 + bias computed once in registers; inner loop streams j rows.
// Non-temporal stores keep the 455 MB output stream from evicting the
// 9.4 MB A-matrix (and bias) out of L2.
// ---------------------------------------------------------------------------
__device__ __forceinline__ float fast_tanh(float x) {
  // tanh(x) = 1 - 2/(exp2(2*log2(e)*x) + 1); exact at +-inf saturation.
  float e = __builtin_amdgcn_exp2f(x * 2.8853900817779268f);
  return 1.0f - 2.0f * __builtin_amdgcn_rcpf(e + 1.0f);
}

__global__ __launch_bounds__(192) void combine_pairs(const float* __restrict__ C,
                                                     const float* __restrict__ bias,
                                                     float* __restrict__ out) {
  const int i  = blockIdx.x;          // row index (0..L-1); biggest blocks first
  const int b  = blockIdx.y;          // batch
  const int h4 = threadIdx.x * 4;     // h vector slot

  const float* a1row = C + (b * Lseq + i) * Ncols + h4;
  v4f base = *(const v4f*)a1row + *(const v4f*)(bias + h4);

  const int off_i = i * (2 * Lseq - i + 1) / 2;       // pairs before row i
  float* orow = out + (b * Pn + off_i) * Hdim + h4;   // max index 113.8M, fits int32
  const float* a2 = C + b * Lseq * Ncols + Hdim + h4; // A2 half, row stride Ncols

  for (int j = i; j < Lseq; ++j) {
    v4f s = base + *(const v4f*)(a2 + j * Ncols);
    v4f r;
    r.x = fast_tanh(s.x);
    r.y = fast_tanh(s.y);
    r.z = fast_tanh(s.z);
    r.w = fast_tanh(s.w);
    __builtin_nontemporal_store(r, (v4f*)orow);
    orow += Hdim;
  }
}

extern "C" void kernel_launch(void* const* d_in, const int* in_sizes, int n_in,
                              void* d_out, int out_size, void* d_ws, size_t ws_size,
                              hipStream_t stream) {
  (void)in_sizes; (void)n_in; (void)out_size; (void)ws_size;
  const float* X    = (const float*)d_in[0];   // seq_hiddens (8,192,768) f32
  const float* W    = (const float*)d_in[1];   // (768, 1536) f32 row-major
  const float* bias = (const float*)d_in[2];   // (768,) f32
  float*       out  = (float*)d_out;           // (8, 18528, 768) f32
  float*       C    = (float*)d_ws;            // 1536 x 1536 f32 = 9.4 MB scratch

  dim3 g1(Ncols / 64, Mrows / 128);            // (24, 12)
  gemm_proj<<<g1, 256, 0, stream>>>(X, W, C);

  dim3 g2(Lseq, Bn);                           // (192, 8)
  combine_pairs<<<g2, 192, 0, stream>>>(C, bias, out);
}